// Routing_7670811590780
// MI455X (gfx1250) — compile-verified
//
#include <hip/hip_runtime.h>

// ---------------------------------------------------------------------------
// Capsule routing step on MI455X (gfx1250).
//   pred = per-group u @ W^T            (WMMA f32 16x16x4)
//   c    = softmax(b, axis=-1)          (VALU + wave shuffles)
//   s    = c^T @ pred   [128 x 64]      (WMMA f32 16x16x4, K=65536 split)
//   v    = squash(s)                    (tiny kernel)
// HBM floor ~40.5 MB / 23.3 TB/s ~= 1.7 us. WMMA keeps the 1.35 GFLOP of
// matrix math off the VALU critical path; TDM (tensor_load_to_lds) moves the
// staging traffic HBM->LDS without a VGPR round-trip.
// ---------------------------------------------------------------------------

typedef float v2f  __attribute__((ext_vector_type(2)));
typedef float v8f  __attribute__((ext_vector_type(8)));
typedef unsigned int u32x4 __attribute__((ext_vector_type(4)));
typedef int   i32x4 __attribute__((ext_vector_type(4)));
typedef int   i32x8 __attribute__((ext_vector_type(8)));

#define IN_DIM   32
#define OUT_DIM  64
#define N_IN     65536
#define N_OUT    128
#define CHUNK    64                 // input capsules per block iteration
#define NCHUNKS  (N_IN / CHUNK)     // 1024
#define THREADS  256                // 8 waves (wave32)
#define NBLOCKS  256                // 4 chunks per block; >=2 blocks/WGP overlap

// LDS partition (single block, constant offsets usable as D#.lds_addr)
#define CB_OFF   0                      // cbuf: 64*128 f32 = 32 KB
#define UB_OFF   (CHUNK * N_OUT)        // ubuf: 64*32  f32 =  8 KB
#define WB_OFF   (UB_OFF + CHUNK * IN_DIM)   // wbuf: 64*32 f32 = 8 KB
#define PB_OFF   (WB_OFF + OUT_DIM * IN_DIM) // pbuf: 64*64 f32 = 16 KB
#define SMEM_F32 (PB_OFF + CHUNK * OUT_DIM)  // 16384 floats = 64 KB

__global__ void zero_ws_kernel(float* ws) {
    int i = blockIdx.x * blockDim.x + threadIdx.x;
    if (i < N_OUT * OUT_DIM) ws[i] = 0.0f;
}

// Tensor Data Mover: 1-D contiguous tile of `nelem` f32, global -> LDS.
// D# layout per CDNA5 ISA ch.8 (group0 128b, group1 256b; groups 2/3 NULL-ish
// zeros for a <=2D tensor). nelem must be <= 65535 (tile_dim0 is 16-bit).
// amdgpu-toolchain (clang-23) 6-arg builtin form.
__device__ __forceinline__ void tdm_load_1d(unsigned lds_byte_off,
                                            const void* gptr, unsigned nelem) {
    unsigned long long ga = (unsigned long long)gptr;
    u32x4 g0;
    g0[0] = 1u;                                   // count=1, user desc
    g0[1] = lds_byte_off;                         // lds_addr (bytes)
    g0[2] = (unsigned)(ga & 0xffffffffu);         // global_addr[31:0]
    g0[3] = (unsigned)((ga >> 32) & 0x01ffffffu)  // global_addr[56:32]
          | 0x80000000u;                          // type=2 ("image")
    i32x8 g1;
    g1[0] = (int)(2u << 16);                      // data_size=4B; mask=0
    g1[1] = (int)((nelem & 0xffffu) << 16);       // tensor_dim0[15:0]
    g1[2] = (int)(((nelem >> 16) & 0xffffu)       // tensor_dim0[31:16]
          | (1u << 16));                          // tensor_dim1 = 1
    g1[3] = (int)((nelem & 0xffffu) << 16);       // tile_dim0 = nelem
    g1[4] = 0;                                    // tile_dim1=0, tile_dim2=0
    g1[5] = (int)nelem;                           // tensor_dim0_stride[31:0]
    g1[6] = 0;                                    // stride hi / dim1_stride lo
    g1[7] = 0;
    i32x4 z4 = {0, 0, 0, 0};
    i32x8 z8 = {0, 0, 0, 0, 0, 0, 0, 0};
    __builtin_amdgcn_tensor_load_to_lds(g0, g1, z4, z4, z8, 0);
}

__global__ __launch_bounds__(THREADS)
void caps_accum_kernel(const float* __restrict__ x,   // u  [65536][32] (contiguous reshape)
                       const float* __restrict__ W,   // W  [64][64][32]
                       const float* __restrict__ b,   // b  [65536][128]
                       float* __restrict__ s_acc)     // s  [128][64] partial accumulator
{
    __shared__ float smem[SMEM_F32];
    float* cbuf = smem + CB_OFF;
    float* ubuf = smem + UB_OFF;
    float* wbuf = smem + WB_OFF;
    float* pbuf = smem + PB_OFF;

    const int t     = threadIdx.x;
    const int lane  = t & 31;
    const int wave  = t >> 5;        // 0..7
    const int l16   = lane & 15;     // column-in-tile
    const int lhalf = lane >> 4;     // 0 or 1 (upper half of wave)

    // Big-GEMM ownership: wave handles N-tile nt2 (of 4 over OUT_DIM) and
    // four M-tiles (of 8 over N_OUT) => 4 persistent 16x16 f32 accumulators.
    const int nt2   = wave & 3;
    const int mtb   = (wave >> 2) * 4;   // M-tile base: 0 or 4
    v8f accD[4] = {};

    for (int chunk = blockIdx.x; chunk < NCHUNKS; chunk += gridDim.x) {
        const int n0 = chunk * CHUNK;
        const int sg = n0 >> 10;            // shared group (chunk never straddles)

        // ---- TDM staging: HBM -> LDS, one wave issues, all waves consume --
        if (wave == 0) {
            tdm_load_1d(CB_OFF * 4u, b + (size_t)n0 * N_OUT, CHUNK * N_OUT);
            tdm_load_1d(UB_OFF * 4u, x + (size_t)n0 * IN_DIM, CHUNK * IN_DIM);
            tdm_load_1d(WB_OFF * 4u, W + (size_t)sg * OUT_DIM * IN_DIM,
                        OUT_DIM * IN_DIM);
            __builtin_amdgcn_s_wait_tensorcnt(0);
        }
        __syncthreads();

        // ---- softmax over each cbuf row (128 wide): 4 lanes per row ------
        {
            const int row = t >> 2;          // 0..63
            const int q   = t & 3;           // quarter of the row
            float* rp = cbuf + row * N_OUT + q * 32;
            float m = -3.402823466e38f;
            #pragma unroll
            for (int i = 0; i < 32; i++) m = fmaxf(m, rp[i]);
            m = fmaxf(m, __shfl_xor(m, 1));
            m = fmaxf(m, __shfl_xor(m, 2));
            float sum = 0.0f;
            #pragma unroll
            for (int i = 0; i < 32; i++) { float e = __expf(rp[i] - m); rp[i] = e; sum += e; }
            sum += __shfl_xor(sum, 1);
            sum += __shfl_xor(sum, 2);
            const float inv = 1.0f / sum;
            #pragma unroll
            for (int i = 0; i < 32; i++) rp[i] *= inv;
        }
        __syncthreads();

        // ---- pred[64n x 64e] = u[64x32] * W^T[32x64] via WMMA ------------
        // 16 tiles (mt,nt); each wave computes 2.
        #pragma unroll
        for (int ti = 0; ti < 2; ti++) {
            const int tile = wave * 2 + ti;
            const int mt = tile >> 2;        // n-tile
            const int nt = tile & 3;         // e-tile
            v8f acc = {};
            #pragma unroll
            for (int kk = 0; kk < 8; kk++) { // K = 32, 4 per WMMA
                const int k0 = kk * 4;
                v2f a;                        // A[m][k] = u[mt*16+m][k0+k]
                a.x = ubuf[(mt * 16 + l16) * IN_DIM + k0 + lhalf * 2];
                a.y = ubuf[(mt * 16 + l16) * IN_DIM + k0 + lhalf * 2 + 1];
                v2f bb;                       // B[k][e] = W[e][k]
                bb.x = wbuf[(nt * 16 + l16) * IN_DIM + k0 + lhalf];
                bb.y = wbuf[(nt * 16 + l16) * IN_DIM + k0 + 2 + lhalf];
                acc = __builtin_amdgcn_wmma_f32_16x16x4_f32(
                        false, a, false, bb, (short)0, acc, false, false);
            }
            #pragma unroll
            for (int r = 0; r < 8; r++)       // D layout: M = r + 8*lhalf, N = l16
                pbuf[(mt * 16 + r + lhalf * 8) * OUT_DIM + nt * 16 + l16] = acc[r];
        }
        __syncthreads();

        // ---- s += c^T[128 x 64] * pred[64 x 64] via WMMA -----------------
        #pragma unroll
        for (int kk = 0; kk < 16; kk++) {     // K = CHUNK = 64, 4 per WMMA
            const int k0 = kk * 4;
            v2f bb;                           // B[k][e] = pred[n][e]
            bb.x = pbuf[(k0 + lhalf)     * OUT_DIM + nt2 * 16 + l16];
            bb.y = pbuf[(k0 + 2 + lhalf) * OUT_DIM + nt2 * 16 + l16];
            #pragma unroll
            for (int i = 0; i < 4; i++) {
                const int mt = mtb + i;
                v2f a;                        // A[o][k] = c[n][o]
                a.x = cbuf[(k0 + lhalf * 2)     * N_OUT + mt * 16 + l16];
                a.y = cbuf[(k0 + lhalf * 2 + 1) * N_OUT + mt * 16 + l16];
                accD[i] = __builtin_amdgcn_wmma_f32_16x16x4_f32(
                            false, a, false, bb, (short)0, accD[i], false, false);
            }
        }
        __syncthreads();   // protect LDS before next chunk's staging
    }

    // ---- reduce partial tiles across blocks --------------------------------
    #pragma unroll
    for (int i = 0; i < 4; i++) {
        const int mt = mtb + i;
        #pragma unroll
        for (int r = 0; r < 8; r++) {
            const int row = mt * 16 + r + lhalf * 8;
            const int col = nt2 * 16 + l16;
            atomicAdd(&s_acc[row * OUT_DIM + col], accD[i][r]);
        }
    }
}

// squash: one wave per output capsule row (128 rows x 64 cols)
__global__ __launch_bounds__(32)
void caps_squash_kernel(const float* __restrict__ s_acc, float* __restrict__ out) {
    const int row  = blockIdx.x;
    const int lane = threadIdx.x;
    const float a0 = s_acc[row * OUT_DIM + lane];
    const float a1 = s_acc[row * OUT_DIM + 32 + lane];
    float n2 = a0 * a0 + a1 * a1;
    #pragma unroll
    for (int off = 16; off > 0; off >>= 1) n2 += __shfl_xor(n2, off);
    const float scale = n2 / ((1.0f + n2) * sqrtf(n2 + 1e-8f));
    out[row * OUT_DIM + lane]      = a0 * scale;
    out[row * OUT_DIM + 32 + lane] = a1 * scale;
}

extern "C" void kernel_launch(void* const* d_in, const int* in_sizes, int n_in,
                              void* d_out, int out_size, void* d_ws, size_t ws_size,
                              hipStream_t stream) {
    const float* x = (const float*)d_in[0];   // [1, 2048, 32, 32] -> u[65536][32]
    const float* W = (const float*)d_in[1];   // [64][64][32]
    const float* b = (const float*)d_in[2];   // [65536][128]
    float* out   = (float*)d_out;             // v [128][64]
    float* s_acc = (float*)d_ws;              // 8192 floats of scratch

    zero_ws_kernel<<<(N_OUT * OUT_DIM + 255) / 256, 256, 0, stream>>>(s_acc);
    caps_accum_kernel<<<NBLOCKS, THREADS, 0, stream>>>(x, W, b, s_acc);
    caps_squash_kernel<<<N_OUT, 32, 0, stream>>>(s_acc, out);
}